// GAT_model_76965813944921
// MI455X (gfx1250) — compile-verified
//
#include <hip/hip_runtime.h>

// ---------------------------------------------------------------------------
// GAT (2 layers, heads=1, edge features) + linear head + softmax for MI455X.
//
//  * a_edge = edge_attr @ (We @ att_e): kills the E x 64 x 128 edge GEMMs.
//  * CSR-by-dst built once, reused by both layers -> no feature atomics.
//  * Node GEMMs use V_WMMA_F32_16X16X4_F32 (wave32, 16x16 tile per wave),
//    compile-time K/NC -> straight-line loads with immediate offsets.
//  * edge_attr streamed with non-temporal loads (read once, 410 MB) so it
//    does not evict the L2-resident node features (51 MB in 192 MB L2).
// ---------------------------------------------------------------------------

typedef __attribute__((ext_vector_type(2))) float v2f;
typedef __attribute__((ext_vector_type(4))) float f4;
typedef __attribute__((ext_vector_type(8))) float v8f;

__device__ __forceinline__ float wave_max32(float v) {
#pragma unroll
  for (int off = 16; off > 0; off >>= 1) v = fmaxf(v, __shfl_xor(v, off, 32));
  return v;
}
__device__ __forceinline__ float wave_sum32(float v) {
#pragma unroll
  for (int off = 16; off > 0; off >>= 1) v += __shfl_xor(v, off, 32);
  return v;
}

// ---------------- v = We @ att_e (two layers), 64 floats each ---------------
__global__ __launch_bounds__(128) void compute_v_kernel(
    const float* __restrict__ We1, const float* __restrict__ ae1,
    const float* __restrict__ We2, const float* __restrict__ ae2,
    float* __restrict__ v1, float* __restrict__ v2) {
  int t = threadIdx.x;
  if (t < 64) {
    float s = 0.f;
    for (int j = 0; j < 128; ++j) s += We1[t * 128 + j] * ae1[j];
    v1[t] = s;
  } else {
    int d = t - 64;
    float s = 0.f;
    for (int j = 0; j < 128; ++j) s += We2[d * 128 + j] * ae2[j];
    v2[d] = s;
  }
}

// ------- a_edge for both layers in one streaming pass over edge_attr --------
__global__ __launch_bounds__(256) void edge_scores_kernel(
    const float* __restrict__ EA, const float* __restrict__ v1,
    const float* __restrict__ v2, float* __restrict__ a1,
    float* __restrict__ a2, int E) {
  __shared__ float sv1[64], sv2[64];
  if (threadIdx.x < 64) {
    sv1[threadIdx.x] = v1[threadIdx.x];
    sv2[threadIdx.x] = v2[threadIdx.x];
  }
  __syncthreads();
  int e = blockIdx.x * blockDim.x + threadIdx.x;
  if (e >= E) return;
  const f4* p = (const f4*)(EA + (unsigned)e * 64u);
  float s1 = 0.f, s2 = 0.f;
#pragma unroll
  for (int i = 0; i < 16; ++i) {
    f4 q = __builtin_nontemporal_load(&p[i]);  // one-shot stream: NT hint
    s1 += q.x * sv1[4 * i] + q.y * sv1[4 * i + 1] + q.z * sv1[4 * i + 2] + q.w * sv1[4 * i + 3];
    s2 += q.x * sv2[4 * i] + q.y * sv2[4 * i + 1] + q.z * sv2[4 * i + 2] + q.w * sv2[4 * i + 3];
  }
  a1[e] = s1;
  a2[e] = s2;
}

// ----------------------------- CSR construction ----------------------------
__global__ __launch_bounds__(256) void count_kernel(const int* __restrict__ dst,
                                                    unsigned* __restrict__ cnt, int E) {
  int e = blockIdx.x * blockDim.x + threadIdx.x;
  if (e < E) atomicAdd(&cnt[dst[e]], 1u);
}

#define SCAN_B 256
__global__ __launch_bounds__(SCAN_B) void scan1_kernel(
    const unsigned* __restrict__ cnt, int* __restrict__ rowp,
    unsigned* __restrict__ bsum, int n) {
  __shared__ unsigned sm[SCAN_B];
  int i = blockIdx.x * SCAN_B + threadIdx.x;
  unsigned v = (i < n) ? cnt[i] : 0u;
  sm[threadIdx.x] = v;
  __syncthreads();
  for (int off = 1; off < SCAN_B; off <<= 1) {
    unsigned t = (threadIdx.x >= (unsigned)off) ? sm[threadIdx.x - off] : 0u;
    __syncthreads();
    sm[threadIdx.x] += t;
    __syncthreads();
  }
  if (i < n) rowp[i + 1] = (int)sm[threadIdx.x];
  if (i == 0) rowp[0] = 0;
  if (threadIdx.x == SCAN_B - 1) bsum[blockIdx.x] = sm[SCAN_B - 1];
}

__global__ __launch_bounds__(1024) void scan2_kernel(unsigned* __restrict__ bsum, int nb) {
  __shared__ unsigned sm[1024];
  unsigned v = (threadIdx.x < (unsigned)nb) ? bsum[threadIdx.x] : 0u;
  sm[threadIdx.x] = v;
  __syncthreads();
  for (int off = 1; off < 1024; off <<= 1) {
    unsigned t = (threadIdx.x >= (unsigned)off) ? sm[threadIdx.x - off] : 0u;
    __syncthreads();
    sm[threadIdx.x] += t;
    __syncthreads();
  }
  if (threadIdx.x < (unsigned)nb) bsum[threadIdx.x] = sm[threadIdx.x] - v;  // exclusive
}

__global__ __launch_bounds__(SCAN_B) void scan3_kernel(int* __restrict__ rowp,
                                                       const unsigned* __restrict__ bsum, int n) {
  int i = blockIdx.x * SCAN_B + threadIdx.x;
  if (i < n) rowp[i + 1] += (int)bsum[blockIdx.x];
}

__global__ __launch_bounds__(256) void scatter_kernel(
    const int* __restrict__ src, const int* __restrict__ dst,
    const int* __restrict__ rowp, unsigned* __restrict__ fill,
    int* __restrict__ col, int* __restrict__ perm, int E) {
  int e = blockIdx.x * blockDim.x + threadIdx.x;
  if (e >= E) return;
  int d = dst[e];
  int p = rowp[d] + (int)atomicAdd(&fill[d], 1u);
  col[p] = src[e];
  perm[p] = e;
}

// -------------- zero-pad Wl [128,20] -> Wlp [128,32] (one shot) ------------
__global__ __launch_bounds__(256) void pad_wl_kernel(const float* __restrict__ Wl,
                                                     float* __restrict__ Wp) {
  int i = blockIdx.x * 256 + threadIdx.x;  // 128*32 entries
  if (i >= 128 * 32) return;
  int r = i >> 5, c = i & 31;
  Wp[i] = (c < 20) ? Wl[r * 20 + c] : 0.f;
}

// ------------------- WMMA f32 GEMM: C[M,NC] = A[M,K] @ B[K,NC] -------------
// One wave computes one 16x16 tile via V_WMMA_F32_16X16X4_F32, K unrolled.
// K and NC are compile-time -> all loads use immediate offsets, no guards.
// Layouts per CDNA5 ISA 7.12.2 (32-bit A 16x4, 32-bit C/D 16x16, wave32).
template <int K, int NC>
__global__ __launch_bounds__(256) void wmma_gemm_f32(
    const float* __restrict__ A, const float* __restrict__ B,
    float* __restrict__ C, int M) {
  constexpr int ntN = NC / 16;
  const int gwave = (int)((blockIdx.x * 256u + threadIdx.x) >> 5);
  const int lane = threadIdx.x & 31;
  const int tM = gwave / ntN;
  const int tN = gwave - tM * ntN;
  if (tM * 16 >= M) return;
  const int mn = lane & 15;  // A-row / B-col / C-col within tile
  const int kh = lane >> 4;  // K-half selector
  const float* __restrict__ arow = A + (unsigned)(tM * 16 + mn) * (unsigned)K + 2u * kh;
  const float* __restrict__ bp = B + (unsigned)(tN * 16 + mn) + (unsigned)(2 * kh) * (unsigned)NC;
  v8f acc = {};
#pragma unroll
  for (int k = 0; k < K; k += 4) {
    v2f a, b;
    a.x = arow[k];          // global_load_b64 (consecutive pair)
    a.y = arow[k + 1];
    b.x = bp[k * NC];       // immediate-offset b32 loads
    b.y = bp[k * NC + NC];
    acc = __builtin_amdgcn_wmma_f32_16x16x4_f32(false, a, false, b, (short)0,
                                                acc, false, false);
  }
  float* __restrict__ cp = C + (unsigned)(tM * 16 + 8 * kh) * (unsigned)NC + (unsigned)(tN * 16 + mn);
#pragma unroll
  for (int r = 0; r < 8; ++r) cp[r * NC] = acc[r];
}

// ----------- per-node attention scores a_src = h.as, a_dst = h.ad ----------
__global__ __launch_bounds__(256) void node_scores_kernel(
    const float* __restrict__ H, const float* __restrict__ sv,
    const float* __restrict__ dv, float* __restrict__ aS,
    float* __restrict__ aD, int N) {
  int node = (int)((blockIdx.x * 256u + threadIdx.x) >> 5);
  int lane = threadIdx.x & 31;
  if (node >= N) return;
  f4 h = *(const f4*)(H + (unsigned)node * 128u + (unsigned)lane * 4u);
  f4 s = *(const f4*)(sv + lane * 4);
  f4 d = *(const f4*)(dv + lane * 4);
  float ps = h.x * s.x + h.y * s.y + h.z * s.z + h.w * s.w;
  float pd = h.x * d.x + h.y * d.y + h.z * d.z + h.w * d.w;
  ps = wave_sum32(ps);
  pd = wave_sum32(pd);
  if (lane == 0) {
    aS[node] = ps;
    aD[node] = pd;
  }
}

// -------- per-dst-node softmax over incoming edges + feature aggregation ---
// One wave per node: register 128-wide accumulator, no atomics.
__global__ __launch_bounds__(256) void aggregate_kernel(
    const float* __restrict__ H, const int* __restrict__ rowp,
    const int* __restrict__ col, const int* __restrict__ perm,
    const float* __restrict__ aE, const float* __restrict__ aS,
    const float* __restrict__ aD, const float* __restrict__ bias,
    float* __restrict__ out, int N) {
  int node = (int)((blockIdx.x * 256u + threadIdx.x) >> 5);
  int lane = threadIdx.x & 31;
  if (node >= N) return;
  int start = rowp[node], end = rowp[node + 1];
  float adn = aD[node];
  f4 acc = {0.f, 0.f, 0.f, 0.f};
  if (end > start) {
    // pass 1: segment max of leaky_relu(alpha)
    float m = -__builtin_inff();
    for (int i = start + lane; i < end; i += 32) {
      float al = aS[col[i]] + adn + aE[perm[i]];
      al = (al > 0.f) ? al : 0.2f * al;
      m = fmaxf(m, al);
    }
    m = wave_max32(m);
    // pass 2: segment sum of exp(alpha - m)
    float s = 0.f;
    for (int i = start + lane; i < end; i += 32) {
      float al = aS[col[i]] + adn + aE[perm[i]];
      al = (al > 0.f) ? al : 0.2f * al;
      s += __expf(al - m);
    }
    s = wave_sum32(s);
    float inv = 1.f / s;
    // pass 3: out = sum_e w_e * h[src_e]  (all lanes share edge e; broadcast
    // scalar loads, each lane owns 4 contiguous feature cols -> b128 gather)
    for (int i = start; i < end; ++i) {
      int cs = col[i];
      float al = aS[cs] + adn + aE[perm[i]];
      al = (al > 0.f) ? al : 0.2f * al;
      float wgt = __expf(al - m) * inv;
      f4 h = *(const f4*)(H + (unsigned)cs * 128u + (unsigned)lane * 4u);
      acc.x += wgt * h.x;
      acc.y += wgt * h.y;
      acc.z += wgt * h.z;
      acc.w += wgt * h.w;
    }
  }
  f4 b = *(const f4*)(bias + lane * 4);
  f4 o = {acc.x + b.x, acc.y + b.y, acc.z + b.z, acc.w + b.w};
  *(f4*)(out + (unsigned)node * 128u + (unsigned)lane * 4u) = o;
}

// ------------- row softmax over 20 classes (logits stored ld=32) -----------
__global__ __launch_bounds__(256) void softmax_kernel(
    const float* __restrict__ logits, const float* __restrict__ bl,
    float* __restrict__ out, int N) {
  int row = (int)((blockIdx.x * 256u + threadIdx.x) >> 5);
  int lane = threadIdx.x & 31;
  if (row >= N) return;
  float v = (lane < 20) ? logits[(unsigned)row * 32u + lane] + bl[lane]
                        : -__builtin_inff();
  float m = wave_max32(v);
  float e = (lane < 20) ? __expf(v - m) : 0.f;
  float s = wave_sum32(e);
  if (lane < 20) out[(unsigned)row * 20u + lane] = e / s;
}

// ---------------------------------------------------------------------------
extern "C" void kernel_launch(void* const* d_in, const int* in_sizes, int n_in,
                              void* d_out, int out_size, void* d_ws, size_t ws_size,
                              hipStream_t stream) {
  const float* x   = (const float*)d_in[0];
  const int*   eix = (const int*)d_in[1];   // int32 per harness dtype rules
  const float* ea  = (const float*)d_in[2];
  const float* W1  = (const float*)d_in[3];
  const float* as1 = (const float*)d_in[4];
  const float* ad1 = (const float*)d_in[5];
  const float* We1 = (const float*)d_in[6];
  const float* ae1 = (const float*)d_in[7];
  const float* b1  = (const float*)d_in[8];
  const float* W2  = (const float*)d_in[9];
  const float* as2 = (const float*)d_in[10];
  const float* ad2 = (const float*)d_in[11];
  const float* We2 = (const float*)d_in[12];
  const float* ae2 = (const float*)d_in[13];
  const float* b2  = (const float*)d_in[14];
  const float* Wl  = (const float*)d_in[15];
  const float* bl  = (const float*)d_in[16];
  float* out = (float*)d_out;
  (void)n_in; (void)out_size; (void)ws_size;

  const int N = in_sizes[0] / 128;
  const int E = in_sizes[2] / 64;
  const int* esrc = eix;
  const int* edst = eix + E;

  // Workspace carve-up (256B aligned). Total ~145 MB.
  char* wp = (char*)d_ws;
  auto take = [&](size_t bytes) {
    void* p = (void*)wp;
    wp += (bytes + 255) & ~(size_t)255;
    return p;
  };
  float*    P      = (float*)take((size_t)N * 128 * 4);
  float*    Q      = (float*)take((size_t)N * 128 * 4);
  float*    aS     = (float*)take((size_t)N * 4);
  float*    aD     = (float*)take((size_t)N * 4);
  float*    ae1e   = (float*)take((size_t)E * 4);
  float*    ae2e   = (float*)take((size_t)E * 4);
  int*      rowp   = (int*)take((size_t)(N + 1) * 4);
  unsigned* cnt    = (unsigned*)take((size_t)N * 4);
  unsigned* fill   = (unsigned*)take((size_t)N * 4);
  int*      col    = (int*)take((size_t)E * 4);
  int*      perm   = (int*)take((size_t)E * 4);
  unsigned* bsum   = (unsigned*)take(4096);
  float*    v1     = (float*)take(512);
  float*    v2     = (float*)take(512);
  float*    Wlp    = (float*)take(128 * 32 * 4);
  float*    logits = (float*)take((size_t)N * 32 * 4);

  const int tb = 256;
  const int egrid = (E + tb - 1) / tb;
  const int ngrid = (N + 7) / 8;            // 8 waves / block, 1 node / wave
  const int nb = (N + tb - 1) / tb;         // scan blocks (<= 1024 required)

  // Edge scores for both layers (single streaming pass over edge_attr).
  compute_v_kernel<<<1, 128, 0, stream>>>(We1, ae1, We2, ae2, v1, v2);
  edge_scores_kernel<<<egrid, tb, 0, stream>>>(ea, v1, v2, ae1e, ae2e, E);

  // CSR by destination (shared by both layers).
  hipMemsetAsync(cnt, 0, (size_t)N * 4, stream);
  count_kernel<<<egrid, tb, 0, stream>>>(edst, cnt, E);
  scan1_kernel<<<nb, tb, 0, stream>>>(cnt, rowp, bsum, N);
  scan2_kernel<<<1, 1024, 0, stream>>>(bsum, nb);
  scan3_kernel<<<nb, tb, 0, stream>>>(rowp, bsum, N);
  hipMemsetAsync(fill, 0, (size_t)N * 4, stream);
  scatter_kernel<<<egrid, tb, 0, stream>>>(esrc, edst, rowp, fill, col, perm, E);

  // Pad classifier weights to 32 columns (removes all GEMM guards).
  pad_wl_kernel<<<16, 256, 0, stream>>>(Wl, Wlp);

  const int mt = (N + 15) / 16;

  // Layer 1: P = x @ W1 ; Q = aggregate(P) + b1
  wmma_gemm_f32<128, 128><<<mt, 256, 0, stream>>>(x, W1, P, N);
  node_scores_kernel<<<ngrid, 256, 0, stream>>>(P, as1, ad1, aS, aD, N);
  aggregate_kernel<<<ngrid, 256, 0, stream>>>(P, rowp, col, perm, ae1e, aS, aD, b1, Q, N);

  // Layer 2: P = Q @ W2 ; Q = aggregate(P) + b2
  wmma_gemm_f32<128, 128><<<mt, 256, 0, stream>>>(Q, W2, P, N);
  node_scores_kernel<<<ngrid, 256, 0, stream>>>(P, as2, ad2, aS, aD, N);
  aggregate_kernel<<<ngrid, 256, 0, stream>>>(P, rowp, col, perm, ae2e, aS, aD, b2, Q, N);

  // Classifier: logits = Q @ Wlp (ld=32) ; out = softmax(logits + bl)
  wmma_gemm_f32<128, 32><<<(mt * 2 + 7) / 8, 256, 0, stream>>>(Q, Wlp, logits, N);
  softmax_kernel<<<ngrid, 256, 0, stream>>>(logits, bl, out, N);
}